// Attention_11879879544368
// MI455X (gfx1250) — compile-verified
//
#include <hip/hip_runtime.h>
#include <hip/hip_bf16.h>

typedef __attribute__((ext_vector_type(2))) float v2f;
typedef __attribute__((ext_vector_type(8))) float v8f;

#define NPAD 304        // 294 rounded up to 19*16
#define NREAL 294
#define HEADS 8
#define DH 32

// Pair-interleaved LDS strides (floats). Each chosen so (stride % 64)==32
// (the two 16-lane halves read adjacent pair-rows -> disjoint bank halves),
// or simply even for already-contiguous k-pairs (Q, P, proj-A).
#define BSTP 224        // w_qkv^T pairs: 128 rows x (96*2 -> 224)
#define KTP  608        // K^T pairs: 16 rows x (304*2)
#define VSP  96         // V pairs: 152 rows x (32*2)
#define QSP  34         // Q row stride (even -> b64-aligned pairs)
#define PSP  18         // P row stride (even)
#define ASP  36         // proj A row stride (mult of 4 -> 16B-aligned b128 dst)
#define BSP2 288        // proj B pairs: 16 rows x (128*2 -> 288), 288%64==32

#define Z8 ((v8f){0.f,0.f,0.f,0.f,0.f,0.f,0.f,0.f})
#define WMMA_F32(a, b, c) \
  __builtin_amdgcn_wmma_f32_16x16x4_f32(false, (a), false, (b), (short)0, (c), false, false)

// ---- CDNA5 async global->LDS staging (falls back to load+ds_store) ----
#if __has_builtin(__builtin_amdgcn_global_load_async_to_lds_b128) && \
    __has_builtin(__builtin_amdgcn_s_wait_asynccnt)
#define USE_ASYNC_LDS 1
typedef __attribute__((__vector_size__(4 * sizeof(int)))) int i4v;
typedef __attribute__((address_space(1))) i4v gi4;   // global int4
typedef __attribute__((address_space(3))) i4v li4;   // LDS int4
#endif

__device__ __forceinline__ void stage16(const float* g, float* l) {
#ifdef USE_ASYNC_LDS
  __builtin_amdgcn_global_load_async_to_lds_b128((gi4*)g, (li4*)l, 0, 0);
#else
  float4 v = *(const float4*)g;
  l[0] = v.x; l[1] = v.y; l[2] = v.z; l[3] = v.w;
#endif
}
__device__ __forceinline__ void stage_wait() {
#ifdef USE_ASYNC_LDS
  __builtin_amdgcn_s_wait_asynccnt(0);
#endif
}

__device__ __forceinline__ v2f lds2(const float* p) { return *(const v2f*)p; }

// ---------------------------------------------------------------------------
// Kernel 0: materialize bias matrix bias[h][i][j] (padded cols -> -1e30)
// ---------------------------------------------------------------------------
__global__ __launch_bounds__(256) void bias_kernel(const float* __restrict__ bias_table,
                                                   const int* __restrict__ rel_idx,
                                                   float* __restrict__ biasM) {
  int idx = blockIdx.x * 256 + threadIdx.x;
  if (idx >= HEADS * NPAD * NPAD) return;
  int h   = idx / (NPAD * NPAD);
  int rem = idx % (NPAD * NPAD);
  int i = rem / NPAD, j = rem % NPAD;
  float v = -1e30f;
  if (i < NREAL && j < NREAL) v = bias_table[rel_idx[i * NREAL + j] * HEADS + h];
  biasM[idx] = v;
}

// ---------------------------------------------------------------------------
// Kernel 1 (fused): QKV projection + flash attention, one WG per (win, head).
// All LDS-sourced WMMA fragments are single aligned ds_load_b64 (pair-interleaved LDS).
// Padded rows clamp to row 0 (finite data, killed by -1e30 bias / zero probs).
// ---------------------------------------------------------------------------
__global__ __launch_bounds__(256) void fused_qkv_attn(const float* __restrict__ x,
                                                      const float* __restrict__ w_qkv,
                                                      const float* __restrict__ biasM,
                                                      float* __restrict__ Ob) {
  extern __shared__ float sm[];
  float* Bs2 = sm;                    // 128 * BSTP : w_qkv^T head block (k-pairs)
  float* Qs  = Bs2 + 128 * BSTP;      // 304 * QSP
  float* Ktp = Qs + 304 * QSP;        // 16 * KTP   (k-pairs of K^T)
  float* Vp  = Ktp + 16 * KTP;        // 152 * VSP  (j-pairs of V)
  float* Ps  = Vp + 152 * VSP;        // 8 * 16 * PSP

  const int win = blockIdx.y, h = blockIdx.x;
  const int tid = threadIdx.x, lane = tid & 31, wid = tid >> 5;
  const int l15   = lane & 15;
  const int half  = lane >> 4;        // 0/1
  const int khalf = half * 2;         // K sub-offset within k-chunk
  const int mhalf = half * 8;         // M offset for C/D rows
  const float scale = 0.17677669529663687f;  // 1/sqrt(32)

  // ---- Stage Bs2[(k/2)][2j + (k&1)] = w_qkv[sect*256 + h*32 + (j&31)][k] ----
  for (int e = tid; e < 96 * 64; e += 256) {
    int j  = e >> 6;            // 0..95
    int c4 = (e & 63) * 4;      // k base 0..252
    int sect = j >> 5;
    int col  = j & 31;
    float4 v = *(const float4*)(w_qkv + (long)(sect * 256 + h * 32 + col) * 256 + c4);
    float* b0 = Bs2 + (c4 >> 1) * BSTP + j * 2;
    b0[0]            = v.x;   // k = c4   (even)
    b0[1]            = v.y;   // k = c4+1
    b0[BSTP + 0]     = v.z;   // k = c4+2
    b0[BSTP + 1]     = v.w;   // k = c4+3
  }
  __syncthreads();

  // ---- Phase 1: per-wave m-tiles of Q/K/V straight into LDS ----
  for (int mt = wid; mt < 19; mt += 8) {
    const int row  = mt * 16 + l15;
    const int rowc = (row < NREAL) ? row : 0;     // clamp padded rows to row 0
    const int li = rowc / 49, rem = rowc % 49;
    const float* ap = x + ((long)(li * 256 + win) * 49 + rem) * 256 + khalf;

    v8f acc[6];
#pragma unroll
    for (int nt = 0; nt < 6; ++nt) acc[nt] = Z8;

    v2f a_cur = *(const v2f*)ap;                  // kk = 0
    for (int kk = 0; kk < 64; ++kk) {
      v2f a_nxt;
      if (kk < 63) a_nxt = *(const v2f*)(ap + (kk + 1) * 4);
      else { a_nxt.x = 0.f; a_nxt.y = 0.f; }
      const int kp = kk * 2 + half;
#pragma unroll
      for (int nt = 0; nt < 6; ++nt) {
        v2f b = lds2(Bs2 + kp * BSTP + (nt * 16 + l15) * 2);
        acc[nt] = WMMA_F32(a_cur, b, acc[nt]);
      }
      a_cur = a_nxt;
    }
    // nt 0,1 -> Q (scaled); 2,3 -> K^T pairs; 4,5 -> V pairs.
#pragma unroll
    for (int nt = 0; nt < 6; ++nt) {
      const int sect = nt >> 1;
      const int col  = (nt & 1) * 16 + l15;
#pragma unroll
      for (int r = 0; r < 8; ++r) {
        const int gm = mt * 16 + r + mhalf;
        const float v = acc[nt][r];
        if (sect == 0)      Qs[gm * QSP + col] = v * scale;
        else if (sect == 1) Ktp[(col >> 1) * KTP + gm * 2 + (col & 1)] = v;
        else                Vp[(gm >> 1) * VSP + col * 2 + (gm & 1)]   = v;
      }
    }
  }
  __syncthreads();

  // ---- Phase 2: flash attention ----
  float* Pw = Ps + wid * 16 * PSP;

  for (int rb = wid; rb < 19; rb += 8) {
    v2f qf[8];
#pragma unroll
    for (int kk = 0; kk < 8; ++kk) {
      int k0 = kk * 4 + khalf;
      qf[kk] = lds2(Qs + (rb * 16 + l15) * QSP + k0);
    }

    v8f o0 = Z8, o1 = Z8;
    float mrun[8], lrun[8];
#pragma unroll
    for (int r = 0; r < 8; ++r) { mrun[r] = -3e38f; lrun[r] = 0.f; }

    for (int cb = 0; cb < 19; ++cb) {
      v8f s = Z8;
#pragma unroll
      for (int kk = 0; kk < 8; ++kk) {
        const int kp = kk * 2 + half;
        v2f b = lds2(Ktp + kp * KTP + (cb * 16 + l15) * 2);
        s = WMMA_F32(qf[kk], b, s);
      }
      int gn = cb * 16 + l15;
      float pv[8], alpha[8];
#pragma unroll
      for (int r = 0; r < 8; ++r) {
        int gm = rb * 16 + r + mhalf;
        float sv = s[r] + biasM[((long)h * NPAD + gm) * NPAD + gn];
        float mx = sv;
        mx = fmaxf(mx, __shfl_xor(mx, 1, 32));
        mx = fmaxf(mx, __shfl_xor(mx, 2, 32));
        mx = fmaxf(mx, __shfl_xor(mx, 4, 32));
        mx = fmaxf(mx, __shfl_xor(mx, 8, 32));
        float mnew = fmaxf(mrun[r], mx);
        float a = __expf(mrun[r] - mnew);
        float p = __expf(sv - mnew);
        float ss = p;
        ss += __shfl_xor(ss, 1, 32);
        ss += __shfl_xor(ss, 2, 32);
        ss += __shfl_xor(ss, 4, 32);
        ss += __shfl_xor(ss, 8, 32);
        lrun[r] = lrun[r] * a + ss;
        mrun[r] = mnew;
        pv[r] = p; alpha[r] = a;
      }
#pragma unroll
      for (int r = 0; r < 8; ++r) { o0[r] *= alpha[r]; o1[r] *= alpha[r]; }
#pragma unroll
      for (int r = 0; r < 8; ++r) Pw[(r + mhalf) * PSP + l15] = pv[r];
#pragma unroll
      for (int kk = 0; kk < 4; ++kk) {
        const int k0 = kk * 4 + khalf;
        const int jp = cb * 8 + kk * 2 + half;   // (cb*16 + k0) / 2
        v2f a2 = lds2(Pw + l15 * PSP + k0);
        v2f b0 = lds2(Vp + jp * VSP + l15 * 2);
        v2f b1 = lds2(Vp + jp * VSP + (16 + l15) * 2);
        o0 = WMMA_F32(a2, b0, o0);
        o1 = WMMA_F32(a2, b1, o1);
      }
    }
#pragma unroll
    for (int r = 0; r < 8; ++r) {
      int gm = rb * 16 + r + mhalf;
      if (gm < NREAL) {
        float inv = 1.f / lrun[r];
        long ob = ((long)win * NPAD + gm) * 256 + h * DH;
        Ob[ob + l15]      = o0[r] * inv;
        Ob[ob + 16 + l15] = o1[r] * inv;
      }
    }
  }
}

// ---------------------------------------------------------------------------
// Kernel 2: output projection + scatter to [b,l,X,Y,w1,w2,d] layout.
// A tiles staged via async global->LDS (16B-aligned dst, stride 36).
// ---------------------------------------------------------------------------
__global__ __launch_bounds__(256) void proj_gemm(const float* __restrict__ Ob,
                                                 const float* __restrict__ w_out,
                                                 float* __restrict__ out) {
  __shared__ float As[64 * ASP];
  __shared__ float Bsp[16 * BSP2];

  const int win    = blockIdx.z;
  const int m_base = blockIdx.y * 64;
  const int j_base = blockIdx.x * 128;
  const int tid  = threadIdx.x;
  const int lane = tid & 31;
  const int wid  = tid >> 5;
  const int wm   = (wid >> 1) * 16;
  const int wn   = (wid & 1) * 64;
  const int l15  = lane & 15;
  const int half = lane >> 4;
  const int khalf = half * 2;
  const int mhalf = half * 8;

  v8f acc[4];
#pragma unroll
  for (int jt = 0; jt < 4; ++jt) acc[jt] = Z8;

  for (int kt = 0; kt < 8; ++kt) {
    const int kb = kt * 32;
#pragma unroll
    for (int i = 0; i < 2; ++i) {
      int e = tid + i * 256;
      int row = e >> 3;
      int c4  = (e & 7) * 4;
      int gm  = m_base + row;
      float* dst = &As[row * ASP + c4];
      if (gm < NREAL) {
        stage16(Ob + ((long)win * NPAD + gm) * 256 + kb + c4, dst);
      } else {
        dst[0] = 0.f; dst[1] = 0.f; dst[2] = 0.f; dst[3] = 0.f;
      }
    }
#pragma unroll
    for (int i = 0; i < 4; ++i) {
      int e = tid + i * 256;
      int j  = e >> 3;
      int c4 = (e & 7) * 4;
      float4 v = *(const float4*)(w_out + (long)(j_base + j) * 256 + kb + c4);
      float* b0 = Bsp + (c4 >> 1) * BSP2 + j * 2;
      b0[0]        = v.x;
      b0[1]        = v.y;
      b0[BSP2 + 0] = v.z;
      b0[BSP2 + 1] = v.w;
    }
    stage_wait();
    __syncthreads();
#pragma unroll
    for (int kk = 0; kk < 8; ++kk) {
      const int k0 = kk * 4 + khalf;
      const int kp = kk * 2 + half;
      v2f a = lds2(As + (wm + l15) * ASP + k0);
#pragma unroll
      for (int jt = 0; jt < 4; ++jt) {
        int n = wn + jt * 16 + l15;
        v2f b = lds2(Bsp + kp * BSP2 + n * 2);
        acc[jt] = WMMA_F32(a, b, acc[jt]);
      }
    }
    __syncthreads();
  }

#pragma unroll
  for (int jt = 0; jt < 4; ++jt) {
    int gn = j_base + wn + jt * 16 + l15;
#pragma unroll
    for (int r = 0; r < 8; ++r) {
      int gm = m_base + wm + r + mhalf;
      if (gm < NREAL) {
        int li = gm / 49, rem = gm % 49;
        out[(((long)(li * 256 + win)) * 49 + rem) * 256 + gn] = acc[jt][r];
      }
    }
  }
}

// ---------------------------------------------------------------------------
extern "C" void kernel_launch(void* const* d_in, const int* in_sizes, int n_in,
                              void* d_out, int out_size, void* d_ws, size_t ws_size,
                              hipStream_t stream) {
  const float* x          = (const float*)d_in[0];
  const float* w_qkv      = (const float*)d_in[1];
  const float* w_out      = (const float*)d_in[2];
  const float* bias_table = (const float*)d_in[3];
  const int*   rel_idx    = (const int*)d_in[4];
  float* out = (float*)d_out;

  float* ws = (float*)d_ws;
  float* Ob    = ws;                                   // 256*304*256 floats
  float* biasM = ws + (long)256 * NPAD * 256;          // 8*304*304 floats

  bias_kernel<<<(HEADS * NPAD * NPAD + 255) / 256, 256, 0, stream>>>(bias_table, rel_idx, biasM);

  size_t fused_lds = (size_t)(128 * BSTP + 304 * QSP + 16 * KTP + 152 * VSP + 8 * 16 * PSP)
                     * sizeof(float);
  fused_qkv_attn<<<dim3(HEADS, 256), 256, fused_lds, stream>>>(x, w_qkv, biasM, Ob);

  proj_gemm<<<dim3(2, 5, 256), 256, 0, stream>>>(Ob, w_out, out);
}